// ReverseLSTMLayer_42941083025996
// MI455X (gfx1250) — compile-verified
//
#include <hip/hip_runtime.h>

// Reversed-time LSTM, T=256, B=64, D=H=1024, fp32.
// Persistent single-launch kernel: 256 timesteps with grid-wide barrier between
// steps. Per step: gates[64,4096] = x_t@W_ih^T + h@W_hh^T + b via
// V_WMMA_F32_16X16X4_F32, K=2048 split across 4 waves, fused LSTM cell epilogue.
//
// Grid: 64 n_tiles (H/16) x 4 m_tiles (B/16) = 256 blocks, 128 threads = 4 waves.
// Wave kq: 0 -> x@W_ih K[0,512), 1 -> x@W_ih K[512,1024),
//          2 -> h@W_hh K[0,512), 3 -> h@W_hh K[512,1024).
// Waves 1..3 dump partials to LDS; wave 0 reduces and runs the cell update.

typedef __attribute__((ext_vector_type(2))) float v2f;
typedef __attribute__((ext_vector_type(8))) float v8f;

#define T_STEPS 256
#define B_SZ    64
#define D_SZ    1024
#define H_SZ    1024
#define NBLK    256   // 64 n_tiles * 4 m_tiles

__device__ __forceinline__ float sigmoid_f(float x) {
    return 1.0f / (1.0f + __expf(-x));
}
__device__ __forceinline__ float tanh_f(float x) {
    return 2.0f / (1.0f + __expf(-2.0f * x)) - 1.0f;
}

__global__ void init_counter_kernel(unsigned* cnt) { *cnt = 0u; }

__global__ __launch_bounds__(128, 1)
void lstm_persistent_kernel(const float* __restrict__ x,     // [T, B, D]
                            const float* __restrict__ h0,    // [B, H]
                            const float* __restrict__ c0,    // [B, H]
                            const float* __restrict__ W_ih,  // [4H, D]
                            const float* __restrict__ W_hh,  // [4H, H]
                            const float* __restrict__ b_ih,  // [4H]
                            const float* __restrict__ b_hh,  // [4H]
                            float* __restrict__ ys,          // [T, B, H]
                            float* __restrict__ hT,          // [B, H]
                            float* __restrict__ cT,          // [B, H]
                            float* __restrict__ c_ws,        // [B, H]
                            unsigned* __restrict__ cnt)      // grid barrier counter
{
    __shared__ float lds_acc[3][4][256];   // [kq-1][gate][lane*8+r] = 12 KB

    const int tid    = threadIdx.x;
    const int lane   = tid & 31;
    const int kq     = tid >> 5;       // 0..3 K-quarter
    const int n_tile = blockIdx.x;     // 0..63
    const int m_tile = blockIdx.y;     // 0..3

    // fp32 16x16x4 fragment layouts (ISA 7.12.2):
    // A: lanes 0-15 -> M=lane, K={k,k+1}; lanes 16-31 -> M=lane-16, K={k+2,k+3}
    // B: lane holds N=lane&15; VGPR0/1 hold K = (lane>>4)*2 + {0,1}
    const int mrow = lane & 15;
    const int koff = (lane >> 4) << 1;           // 0 or 2
    const int ncol = lane & 15;
    const int jcol = n_tile * 16 + ncol;

    const float* W      = (kq < 2) ? W_ih : W_hh;       // both row stride 1024
    const int    kbase  = (kq & 1) * 512;
    const float* w_ptr  = W + (size_t)jcol * 1024 + kbase + koff;
    const size_t gs     = (size_t)H_SZ * 1024;          // W row-block stride per gate

    // Per-column biases (used by wave 0 only; cheap scalar loads)
    const float bi = b_ih[jcol]            + b_hh[jcol];
    const float bf = b_ih[H_SZ + jcol]     + b_hh[H_SZ + jcol];
    const float bg = b_ih[2 * H_SZ + jcol] + b_hh[2 * H_SZ + jcol];
    const float bo = b_ih[3 * H_SZ + jcol] + b_hh[3 * H_SZ + jcol];

    const int row_base = m_tile * 16 + (lane >> 4) * 8;  // C/D row for VGPR r: row_base + r

    for (int s = 0; s < T_STEPS; ++s) {
        const int t = T_STEPS - 1 - s;
        const float* x_t    = x + (size_t)t * B_SZ * D_SZ;
        const float* h_prev = (t == T_STEPS - 1) ? h0 : ys + (size_t)(t + 1) * B_SZ * H_SZ;
        const float* A      = (kq < 2) ? x_t : h_prev;
        const float* a_ptr  = A + (size_t)(m_tile * 16 + mrow) * 1024 + kbase + koff;

        v8f acc0 = {}; v8f acc1 = {}; v8f acc2 = {}; v8f acc3 = {};

        #pragma unroll 4
        for (int k = 0; k < 512; k += 4) {
            v2f a  = *(const v2f*)(a_ptr + k);
            v2f b0 = *(const v2f*)(w_ptr + k);
            v2f b1 = *(const v2f*)(w_ptr + gs     + k);
            v2f b2 = *(const v2f*)(w_ptr + 2 * gs + k);
            v2f b3 = *(const v2f*)(w_ptr + 3 * gs + k);
            acc0 = __builtin_amdgcn_wmma_f32_16x16x4_f32(false, a, false, b0, (short)0, acc0, false, false);
            acc1 = __builtin_amdgcn_wmma_f32_16x16x4_f32(false, a, false, b1, (short)0, acc1, false, false);
            acc2 = __builtin_amdgcn_wmma_f32_16x16x4_f32(false, a, false, b2, (short)0, acc2, false, false);
            acc3 = __builtin_amdgcn_wmma_f32_16x16x4_f32(false, a, false, b3, (short)0, acc3, false, false);
        }

        // K-split reduction: waves 1..3 dump partials, wave 0 combines.
        if (kq != 0) {
            float* sp = &lds_acc[kq - 1][0][lane * 8];
            #pragma unroll
            for (int r = 0; r < 8; ++r) {
                sp[0 * 256 + r] = acc0[r];
                sp[1 * 256 + r] = acc1[r];
                sp[2 * 256 + r] = acc2[r];
                sp[3 * 256 + r] = acc3[r];
            }
        }
        __syncthreads();

        if (kq == 0) {
            #pragma unroll
            for (int q = 0; q < 3; ++q) {
                const float* sp = &lds_acc[q][0][lane * 8];
                #pragma unroll
                for (int r = 0; r < 8; ++r) {
                    acc0[r] += sp[0 * 256 + r];
                    acc1[r] += sp[1 * 256 + r];
                    acc2[r] += sp[2 * 256 + r];
                    acc3[r] += sp[3 * 256 + r];
                }
            }

            const float* c_prev = (t == T_STEPS - 1) ? c0 : c_ws;
            float* h_out = ys + (size_t)t * B_SZ * H_SZ;

            #pragma unroll
            for (int r = 0; r < 8; ++r) {
                const int b_idx  = row_base + r;
                const size_t off = (size_t)b_idx * H_SZ + jcol;
                const float iv = sigmoid_f(acc0[r] + bi);
                const float fv = sigmoid_f(acc1[r] + bf);
                const float gv = tanh_f   (acc2[r] + bg);
                const float ov = sigmoid_f(acc3[r] + bo);
                const float cn = fv * c_prev[off] + iv * gv;
                const float hn = ov * tanh_f(cn);
                c_ws[off]  = cn;
                h_out[off] = hn;
            }
            if (t == 0) {
                #pragma unroll
                for (int r = 0; r < 8; ++r) {
                    const int b_idx  = row_base + r;
                    const size_t off = (size_t)b_idx * H_SZ + jcol;
                    hT[off] = ys[off];     // ys[0*B*H + off]
                    cT[off] = c_ws[off];
                }
            }
        }

        // Grid-wide barrier between timesteps (monotonic counter, no reset race).
        if (s != T_STEPS - 1) {
            __threadfence();              // release h/c stores to device scope
            __syncthreads();              // whole block done (incl. LDS consumers)
            if (tid == 0) {
                __hip_atomic_fetch_add(cnt, 1u, __ATOMIC_ACQ_REL, __HIP_MEMORY_SCOPE_AGENT);
                const unsigned target = (unsigned)NBLK * (unsigned)(s + 1);
                while (__hip_atomic_load(cnt, __ATOMIC_ACQUIRE, __HIP_MEMORY_SCOPE_AGENT) < target) {
                    __builtin_amdgcn_s_sleep(1);
                }
            }
            __syncthreads();
            __threadfence();              // acquire: invalidate before reading peers' h
        }
    }
}

extern "C" void kernel_launch(void* const* d_in, const int* in_sizes, int n_in,
                              void* d_out, int out_size, void* d_ws, size_t ws_size,
                              hipStream_t stream) {
    (void)in_sizes; (void)n_in; (void)out_size; (void)ws_size;
    const float* x    = (const float*)d_in[0];   // [T, B, D]
    const float* h0   = (const float*)d_in[1];   // [B, H]
    const float* c0   = (const float*)d_in[2];   // [B, H]
    const float* W_ih = (const float*)d_in[3];   // [4H, D]
    const float* W_hh = (const float*)d_in[4];   // [4H, H]
    const float* b_ih = (const float*)d_in[5];   // [4H]
    const float* b_hh = (const float*)d_in[6];   // [4H]

    float* out  = (float*)d_out;
    float* ys   = out;                                     // [T, B, H]
    float* hT   = out + (size_t)T_STEPS * B_SZ * H_SZ;     // [B, H]
    float* cT   = hT + (size_t)B_SZ * H_SZ;                // [B, H]

    float*    c_ws = (float*)d_ws;                                          // [B, H]
    unsigned* cnt  = (unsigned*)((char*)d_ws + (size_t)B_SZ * H_SZ * sizeof(float));

    init_counter_kernel<<<1, 1, 0, stream>>>(cnt);

    dim3 grid(H_SZ / 16, B_SZ / 16);   // 64 x 4 = 256 blocks
    dim3 block(128);                   // 4 waves (K-quarters)
    lstm_persistent_kernel<<<grid, block, 0, stream>>>(
        x, h0, c0, W_ih, W_hh, b_ih, b_hh, ys, hT, cT, c_ws, cnt);
}